// VectorQuantizer_27152783245576
// MI455X (gfx1250) — compile-verified
//
#include <hip/hip_runtime.h>
#include <math.h>

typedef __attribute__((ext_vector_type(2))) float v2f;
typedef __attribute__((ext_vector_type(4))) float v4f;
typedef __attribute__((ext_vector_type(8))) float v8f;
typedef __attribute__((ext_vector_type(4))) unsigned int u32x4;
typedef __attribute__((ext_vector_type(8))) int i32x8;
typedef __attribute__((ext_vector_type(4))) int i32x4;

#define K_CB   1024
#define C_DIM  64
#define S_DIM  16384            // D*H*W = 16*32*32
#define B_DIM  4
#define N_TOK  65536            // B*S
#define NC_OUT 4194304          // N_TOK * C_DIM

#define CHROWS 64               // codebook rows per TDM chunk
#define NCHUNK (K_CB / CHROWS)  // 16 chunks
#define LDSTR  68               // padded LDS row stride (floats)

// workspace layout (element offsets, 4-byte units)
#define WS_IDX    0             // int  [65536]
#define WS_CNT    65536         // int  [1024]
#define WS_ENORM  66560         // float[1024]
#define WS_ESUM   67584         // float[64]
#define WS_ZSUM   67648         // float[64]
#define WS_Z2SUM  67712         // float[64]
#define WS_SSE    67776         // float[256]

#if defined(__has_builtin)
#  if __has_builtin(__builtin_amdgcn_tensor_load_to_lds) && \
      __has_builtin(__builtin_amdgcn_s_wait_tensorcnt)
#    define HAVE_TDM 1
#  endif
#endif
#ifndef HAVE_TDM
#  define HAVE_TDM 0
#endif

// ---------------------------------------------------------------- utilities
__global__ void k_zero_counts(int* __restrict__ cnt) {
    int t = blockIdx.x * blockDim.x + threadIdx.x;
    if (t < K_CB) cnt[t] = 0;
}

// enorm[k] = ||E_k||^2
__global__ void k_enorm(const float* __restrict__ E, float* __restrict__ enorm) {
    int k = blockIdx.x * blockDim.x + threadIdx.x;
    if (k >= K_CB) return;
    const float* p = E + (size_t)k * C_DIM;
    float s = 0.f;
    #pragma unroll 8
    for (int c = 0; c < C_DIM; ++c) { float v = p[c]; s += v * v; }
    enorm[k] = s;
}

// esum[c] = sum_k E[k][c]
__global__ void k_esum(const float* __restrict__ E, float* __restrict__ esum) {
    int c = threadIdx.x;
    if (c >= C_DIM) return;
    float s = 0.f;
    for (int k = 0; k < K_CB; ++k) s += E[(size_t)k * C_DIM + c];
    esum[c] = s;
}

// per-channel sums of z and z^2 ; one workgroup per channel (deterministic)
__global__ __launch_bounds__(256) void k_zstats(const float* __restrict__ z,
                                                float* __restrict__ zsum,
                                                float* __restrict__ z2sum) {
    int c = blockIdx.x;
    int t = threadIdx.x;
    float s = 0.f, s2 = 0.f;
    for (int b = 0; b < B_DIM; ++b) {
        const float* p = z + (size_t)b * C_DIM * S_DIM + (size_t)c * S_DIM;
        for (int i = t; i < S_DIM; i += 256) { float v = p[i]; s += v; s2 += v * v; }
    }
    __shared__ float sh[256], sh2[256];
    sh[t] = s; sh2[t] = s2;
    __syncthreads();
    for (int o = 128; o > 0; o >>= 1) {
        if (t < o) { sh[t] += sh[t + o]; sh2[t] += sh2[t + o]; }
        __syncthreads();
    }
    if (t == 0) { zsum[c] = sh[0]; z2sum[c] = sh2[0]; }
}

// ------------------------------------------- WMMA distance GEMM + row argmin
// 8 waves/WG; each wave owns 16 tokens (rows). Codebook streamed through a
// double-buffered TDM pipeline: tensor_load_to_lds for chunk c+1 overlaps the
// WMMA work on chunk c (s_wait_tensorcnt 1 retires only the older DMA).
// TDM pad_interval=64 DWORDs + pad_amount=4 DWORDs produces the 68-float
// padded LDS row stride in hardware -> conflict-free strided B reads.
__global__ __launch_bounds__(256) void k_argmin(const float* __restrict__ z,
                                                const float* __restrict__ E,
                                                const float* __restrict__ enorm,
                                                int* __restrict__ idx_out) {
    __shared__ float lse[2][CHROWS * LDSTR];   // 2 x 17408 B ping-pong
    __shared__ float len[2][CHROWS];           // enorm chunks

    const int tid  = threadIdx.x;
    const int lane = tid & 31;
    const int wave = tid >> 5;
    const int l16  = lane & 15;
    const int hi   = lane >> 4;             // 0 | 1 (half-wave)
    const int koff = hi * 2;                // f32 WMMA K sub-offset

#if HAVE_TDM
    auto tdm_issue = [&](int c, int buf) {
        unsigned long long ga =
            (unsigned long long)(uintptr_t)(E + (size_t)c * CHROWS * C_DIM);
        unsigned lds_off = (unsigned)(uintptr_t)&lse[buf][0];
        u32x4 g0;
        g0[0] = 1u;                                   // count=1, user D#
        g0[1] = lds_off;                              // lds_addr (bytes)
        g0[2] = (unsigned)(ga & 0xFFFFFFFFu);         // global_addr[31:0]
        g0[3] = (unsigned)((ga >> 32) & 0x01FFFFFFu)  // global_addr[56:32]
                | (2u << 30);                         // type = 2 (image)
        i32x8 g1;
        g1[0] = (2 << 16)      // data_size = 4B
              | (1 << 20)      // pad_enable
              | (5 << 22)      // pad_interval: 64 DWORDs
              | (3 << 25);     // pad_amount:   4 DWORDs
        g1[1] = (C_DIM  & 0xFFFF) << 16;  // tensor_dim0 = 64   [63:48]
        g1[2] = (K_CB   & 0xFFFF) << 16;  // tensor_dim1 lo16   [95:80]
        g1[3] = (C_DIM  & 0xFFFF) << 16;  // tile_dim0 = 64     [127:112]
        g1[4] = CHROWS;                   // tile_dim1 = 64     [143:128]
        g1[5] = C_DIM;                    // tensor_dim0_stride [191:160]
        g1[6] = 0;
        g1[7] = 0;
        i32x4 gz = {0, 0, 0, 0};
#if __clang_major__ >= 23
        i32x8 gz8 = {0, 0, 0, 0, 0, 0, 0, 0};
        __builtin_amdgcn_tensor_load_to_lds(g0, g1, gz, gz, gz8, 0);
#else
        __builtin_amdgcn_tensor_load_to_lds(g0, g1, gz, gz, 0);
#endif
    };
#endif

    // ---- A tile: 16 tokens x 64 channels, resident in 32 VGPRs per lane ----
    const int tok0 = blockIdx.x * 128 + wave * 16;
    const int row  = tok0 + l16;
    const int b    = row >> 14;             // /S_DIM (tiles never cross batch)
    const int sp   = row & (S_DIM - 1);
    const float* pz = z + (size_t)b * C_DIM * S_DIM + sp;

    v2f areg[16];
    #pragma unroll
    for (int kc = 0; kc < 16; ++kc) {
        int c0 = 4 * kc + koff;             // lane holds A[M][c0], A[M][c0+1]
        areg[kc].x = pz[(size_t)c0 * S_DIM];
        areg[kc].y = pz[(size_t)(c0 + 1) * S_DIM];
    }

    float bestS[8];
    int   bestI[8];
    #pragma unroll
    for (int v = 0; v < 8; ++v) { bestS[v] = 3.0e38f; bestI[v] = 0; }

#if HAVE_TDM
    if (wave == 0) tdm_issue(0, 0);         // prime the pipeline
#endif

    for (int chunk = 0; chunk < NCHUNK; ++chunk) {
        const int cb = chunk & 1;
        __syncthreads();                    // buffer cb fully consumed (WAR)
#if HAVE_TDM
        if (wave == 0) {
            if (chunk < NCHUNK - 1) {
                tdm_issue(chunk + 1, cb ^ 1);            // prefetch next chunk
                __builtin_amdgcn_s_wait_tensorcnt(1);    // retire current DMA
            } else {
                __builtin_amdgcn_s_wait_tensorcnt(0);
            }
        }
#else
        {   // fallback: cooperative float4 staging of current chunk
            const float* src = E + (size_t)chunk * CHROWS * C_DIM;
            for (int q = tid; q < CHROWS * 16; q += 256) {
                int r  = q >> 4;
                int cq = (q & 15) << 2;
                v4f val = *(const v4f*)(src + r * C_DIM + cq);
                *(v4f*)(&lse[cb][r * LDSTR + cq]) = val;
            }
        }
#endif
        if (tid < CHROWS) len[cb][tid] = enorm[chunk * CHROWS + tid];
        __syncthreads();

        const float* lbase = &lse[cb][0];
        for (int t = 0; t < CHROWS / 16; ++t) {       // N-tiles of 16 cols
            const int nl = t * 16 + l16;              // B col for this lane
            const float* pb = lbase + nl * LDSTR + koff;
            // preload all B fragments -> batched ds_loads
            v2f bv[16];
            #pragma unroll
            for (int kc = 0; kc < 16; ++kc) bv[kc] = *(const v2f*)(pb + 4 * kc);
            // two independent accumulation chains for matrix-pipe ILP
            v8f acc0 = {0.f,0.f,0.f,0.f,0.f,0.f,0.f,0.f};
            v8f acc1 = {0.f,0.f,0.f,0.f,0.f,0.f,0.f,0.f};
            #pragma unroll
            for (int kc = 0; kc < 16; kc += 2) {
                acc0 = __builtin_amdgcn_wmma_f32_16x16x4_f32(
                        false, areg[kc],     false, bv[kc],     (short)0, acc0, false, false);
                acc1 = __builtin_amdgcn_wmma_f32_16x16x4_f32(
                        false, areg[kc + 1], false, bv[kc + 1], (short)0, acc1, false, false);
            }
            const int   nglob = chunk * CHROWS + nl;
            const float en    = len[cb][nl];
            #pragma unroll
            for (int v = 0; v < 8; ++v) {             // M = v + 8*hi, N = nl
                float sc = en - 2.0f * (acc0[v] + acc1[v]);
                if (sc < bestS[v]) { bestS[v] = sc; bestI[v] = nglob; }
            }
        }
    }

    // min-reduce across the 16 lanes of each half-wave (wave32 C/D layout)
    #pragma unroll
    for (int v = 0; v < 8; ++v) {
        float s = bestS[v];
        int   i = bestI[v];
        #pragma unroll
        for (int m = 1; m < 16; m <<= 1) {
            float so = __shfl_xor(s, m, 32);
            int   io = __shfl_xor(i, m, 32);
            if (so < s || (so == s && io < i)) { s = so; i = io; }
        }
        if (l16 == 0) idx_out[tok0 + v + 8 * hi] = i;   // row M = v + 8*hi
    }
}

// ------------------------- gather z_q, write channel-first out, SSE + counts
__global__ __launch_bounds__(256) void k_gather(const float* __restrict__ z,
                                                const float* __restrict__ E,
                                                const int* __restrict__ idx,
                                                float* __restrict__ out,
                                                int* __restrict__ counts,
                                                float* __restrict__ ssep) {
    int i  = blockIdx.x * 256 + threadIdx.x;     // token id
    int b  = i >> 14;
    int sp = i & (S_DIM - 1);
    const float* pz = z + (size_t)b * C_DIM * S_DIM + sp;
    float*       po = out + (size_t)b * C_DIM * S_DIM + sp;
    int e = idx[i];
    const float* pe = E + (size_t)e * C_DIM;
    float sse = 0.f;
    #pragma unroll 8
    for (int c = 0; c < C_DIM; ++c) {
        float q  = pe[c];
        float zv = pz[(size_t)c * S_DIM];
        po[(size_t)c * S_DIM] = q;               // coalesced across threads
        float d = q - zv;
        sse += d * d;
    }
    atomicAdd(&counts[e], 1);                    // integer: deterministic
    __shared__ float sh[256];
    sh[threadIdx.x] = sse;
    __syncthreads();
    for (int o = 128; o > 0; o >>= 1) {
        if (threadIdx.x < o) sh[threadIdx.x] += sh[threadIdx.x + o];
        __syncthreads();
    }
    if (threadIdx.x == 0) ssep[blockIdx.x] = sh[0];
}

// ------------------------------------------------------------ scalar outputs
__global__ __launch_bounds__(256) void k_finalize(const float* __restrict__ enorm,
                                                  const float* __restrict__ esum,
                                                  const float* __restrict__ zsum,
                                                  const float* __restrict__ z2sum,
                                                  const float* __restrict__ ssep,
                                                  const int*   __restrict__ counts,
                                                  float* __restrict__ outs) {
    __shared__ float sh[256];
    int t = threadIdx.x;

    sh[t] = ssep[t];                                    // 256 partials
    __syncthreads();
    for (int o = 128; o > 0; o >>= 1) { if (t < o) sh[t] += sh[t + o]; __syncthreads(); }
    float SSE = sh[0]; __syncthreads();

    float s = 0.f;
    for (int k = t; k < K_CB; k += 256) s += enorm[k];  // Σ||e_k||^2
    sh[t] = s; __syncthreads();
    for (int o = 128; o > 0; o >>= 1) { if (t < o) sh[t] += sh[t + o]; __syncthreads(); }
    float Se2 = sh[0]; __syncthreads();

    s = 0.f; float s2 = 0.f;
    if (t < C_DIM) { s = esum[t] * zsum[t]; s2 = z2sum[t]; }
    sh[t] = s; __syncthreads();
    for (int o = 128; o > 0; o >>= 1) { if (t < o) sh[t] += sh[t + o]; __syncthreads(); }
    float DOT = sh[0]; __syncthreads();
    sh[t] = s2; __syncthreads();
    for (int o = 128; o > 0; o >>= 1) { if (t < o) sh[t] += sh[t + o]; __syncthreads(); }
    float Sz2 = sh[0]; __syncthreads();

    s = 0.f;
    for (int k = t; k < K_CB; k += 256) {
        float em = (float)counts[k] * (1.0f / 65536.0f);
        s += em * logf(em + 1e-10f);
    }
    sh[t] = s; __syncthreads();
    for (int o = 128; o > 0; o >>= 1) { if (t < o) sh[t] += sh[t + o]; __syncthreads(); }
    float H = sh[0];

    if (t == 0) {
        outs[0] = 1.25f * SSE * (1.0f / 4194304.0f);               // loss
        outs[1] = expf(-H);                                        // perplexity
        outs[2] = Sz2 * (1.0f / 65536.0f) + Se2 * (1.0f / 1024.0f)
                  - 2.0f * DOT * (1.0f / 67108864.0f);             // mean(d)
    }
}

// ---------------------------------------------------------------------------
extern "C" void kernel_launch(void* const* d_in, const int* in_sizes, int n_in,
                              void* d_out, int out_size, void* d_ws, size_t ws_size,
                              hipStream_t stream) {
    const float* z = (const float*)d_in[0];   // [4,64,16,32,32]
    const float* E = (const float*)d_in[1];   // [1024,64]
    float* out = (float*)d_out;               // [4,64,16,32,32] ++ 3 scalars

    int*   wsi = (int*)d_ws;
    float* wsf = (float*)d_ws;
    int*   idx    = wsi + WS_IDX;
    int*   counts = wsi + WS_CNT;
    float* enorm  = wsf + WS_ENORM;
    float* esum   = wsf + WS_ESUM;
    float* zsum   = wsf + WS_ZSUM;
    float* z2sum  = wsf + WS_Z2SUM;
    float* ssep   = wsf + WS_SSE;

    k_zero_counts<<<4,   256, 0, stream>>>(counts);
    k_enorm      <<<4,   256, 0, stream>>>(E, enorm);
    k_esum       <<<1,    64, 0, stream>>>(E, esum);
    k_zstats     <<<64,  256, 0, stream>>>(z, zsum, z2sum);
    k_argmin     <<<512, 256, 0, stream>>>(z, E, enorm, idx);
    k_gather     <<<256, 256, 0, stream>>>(z, E, idx, out, counts, ssep);
    k_finalize   <<<1,   256, 0, stream>>>(enorm, esum, zsum, z2sum, ssep, counts,
                                           out + NC_OUT);
}